// PIANO_EnergyV_9655086481804
// MI455X (gfx1250) — compile-verified
//
#include <hip/hip_runtime.h>
#include <hip/hip_bf16.h>

// PIANO EnergyV: 3D anisotropic diffusion + upwind advection stencil.
// Bandwidth-bound fp32 stencil -> stage tiles in LDS via CDNA5 async
// global->LDS DMA (ASYNCcnt) and serve neighbor reads from ds_load.

#define NXX 160
#define NB  2
#define TX  4
#define TY  4
#define TZ  32
#define HXD (TX + 2)
#define HYD (TY + 2)
#define HZD (TZ + 2)
#define HVOL (HXD * HYD * HZD)   // 1224 halo-tile elements per field
#define STRX (HYD * HZD)         // 204
#define STRY (HZD)               // 34
#define STRZ 1
#define NTHREADS 256

#if defined(__has_builtin)
#  if __has_builtin(__builtin_amdgcn_global_load_async_to_lds_b32)
#    define HAVE_ASYNC_LDS 1
#  else
#    define HAVE_ASYNC_LDS 0
#  endif
#  if __has_builtin(__builtin_amdgcn_s_wait_asynccnt)
#    define HAVE_WAIT_ASYNC 1
#  else
#    define HAVE_WAIT_ASYNC 0
#  endif
#else
#  define HAVE_ASYNC_LDS 0
#  define HAVE_WAIT_ASYNC 0
#endif

typedef __attribute__((address_space(1))) int  g_int;
typedef __attribute__((address_space(3))) int  l_int;

__global__ __launch_bounds__(NTHREADS) void piano_energyv_kernel(
    const float* __restrict__ C,   const float* __restrict__ Vx,
    const float* __restrict__ Vy,  const float* __restrict__ Vz,
    const float* __restrict__ Dxx, const float* __restrict__ Dxy,
    const float* __restrict__ Dxz, const float* __restrict__ Dyy,
    const float* __restrict__ Dyz, const float* __restrict__ Dzz,
    float* __restrict__ out)
{
    __shared__ float S[10][HVOL];   // 48.96 KB (of 320 KB/WGP)

    const int tid = threadIdx.x;
    const int x0  = blockIdx.x * TX;
    const int y0  = blockIdx.y * TY;
    const int zb  = blockIdx.z % (NXX / TZ);
    const int b   = blockIdx.z / (NXX / TZ);
    const int z0  = zb * TZ;
    const int batchOff = b * NXX * NXX * NXX;

    const float* const ins[10] = {C, Vx, Vy, Vz, Dxx, Dxy, Dxz, Dyy, Dyz, Dzz};

    // ---- stage halo tiles of all 10 fields into LDS (clamped addressing) ----
#pragma unroll
    for (int f = 0; f < 10; ++f) {
        const float* src = ins[f] + batchOff;
        for (unsigned e = (unsigned)tid; e < HVOL; e += NTHREADS) {
            unsigned hx = e / STRX;
            unsigned r  = e - hx * STRX;
            unsigned hy = r / STRY;
            unsigned hz = r - hy * STRY;
            int gx = min(max(x0 - 1 + (int)hx, 0), NXX - 1);
            int gy = min(max(y0 - 1 + (int)hy, 0), NXX - 1);
            int gz = min(max(z0 - 1 + (int)hz, 0), NXX - 1);
            int gidx = (gx * NXX + gy) * NXX + gz;
#if HAVE_ASYNC_LDS
            __builtin_amdgcn_global_load_async_to_lds_b32(
                (g_int*)(src + gidx),
                (l_int*)&S[f][e],
                /*offset=*/0, /*cpol=*/0);
#else
            S[f][e] = src[gidx];
#endif
        }
    }
#if HAVE_WAIT_ASYNC
    __builtin_amdgcn_s_wait_asynccnt(0);
#elif HAVE_ASYNC_LDS
    asm volatile("s_wait_asynccnt 0" ::: "memory");
#endif
    __syncthreads();

    // ---- per-voxel stencil ----
    const int lz  = tid & (TZ - 1);
    const int ly  = (tid >> 5) & (TY - 1);
    const int lxh = tid >> 7;           // 0 or 1; each thread does 2 x-slices

    const int iy = y0 + ly;
    const int iz = z0 + lz;
    const bool ylo = (iy == 0), yhi = (iy == NXX - 1);
    const bool zlo = (iz == 0), zhi = (iz == NXX - 1);

#pragma unroll
    for (int sub = 0; sub < 2; ++sub) {
        const int lx = lxh * 2 + sub;
        const int ix = x0 + lx;
        const bool xlo = (ix == 0), xhi = (ix == NXX - 1);
        const int h = ((lx + 1) * HYD + (ly + 1)) * HZD + (lz + 1);

        // central diff with one-sided boundaries: 0.5*(fwd+bwd)
        auto cen = [&](int f, int s, bool lo, bool hi) -> float {
            float fm = S[f][h - s], f0 = S[f][h], fp = S[f][h + s];
            float fw = hi ? (f0 - fm) : (fp - f0);
            float bw = lo ? (fp - f0) : (f0 - fm);
            return 0.5f * (fw + bw);
        };
        auto fwdd = [&](int f, int s, bool hi) -> float {
            return hi ? (S[f][h] - S[f][h - s]) : (S[f][h + s] - S[f][h]);
        };
        auto bwdd = [&](int f, int s, bool lo) -> float {
            return lo ? (S[f][h + s] - S[f][h]) : (S[f][h] - S[f][h - s]);
        };
        // bwd(fwd) second derivative of C along one axis
        auto sder = [&](int s, bool lo, bool hi) -> float {
            if (hi) return 0.0f;                       // fwd copies last diff
            int bb = lo ? (h + s) : h;                 // shift one in at i==0
            return S[0][bb + s] - 2.0f * S[0][bb] + S[0][bb - s];
        };
        // M_ab = bwd_a(fwd_b(C))
        auto cross = [&](int sa, bool alo, int sb, bool bhi) -> float {
            auto g = [&](int off) -> float {
                return bhi ? (S[0][h + off] - S[0][h + off - sb])
                           : (S[0][h + off + sb] - S[0][h + off]);
            };
            return alo ? (g(sa) - g(0)) : (g(0) - g(-sa));
        };

        float cxC = cen(0, STRX, xlo, xhi);
        float cyC = cen(0, STRY, ylo, yhi);
        float czC = cen(0, STRZ, zlo, zhi);

        float t1 = cen(4, STRX, xlo, xhi) + cen(5, STRY, ylo, yhi) + cen(6, STRZ, zlo, zhi);
        float t2 = cen(5, STRX, xlo, xhi) + cen(7, STRY, ylo, yhi) + cen(8, STRZ, zlo, zhi);
        float t3 = cen(6, STRX, xlo, xhi) + cen(8, STRY, ylo, yhi) + cen(9, STRZ, zlo, zhi);

        float dxx = S[4][h], dxy = S[5][h], dxz = S[6][h];
        float dyy = S[7][h], dyz = S[8][h], dzz = S[9][h];

        float diff = t1 * cxC + t2 * cyC + t3 * czC
                   + dxx * sder(STRX, xlo, xhi)
                   + dyy * sder(STRY, ylo, yhi)
                   + dzz * sder(STRZ, zlo, zhi)
                   + dxy * (cross(STRX, xlo, STRY, yhi) + cross(STRY, ylo, STRX, xhi))
                   + dyz * (cross(STRY, ylo, STRZ, zhi) + cross(STRZ, zlo, STRY, yhi))
                   + dxz * (cross(STRZ, zlo, STRX, xhi) + cross(STRX, xlo, STRZ, zhi));

        float vx = S[1][h], vy = S[2][h], vz = S[3][h];
        float Cxu = (vx > 0.0f) ? bwdd(0, STRX, xlo) : fwdd(0, STRX, xhi);
        float Cyu = (vy > 0.0f) ? bwdd(0, STRY, ylo) : fwdd(0, STRY, yhi);
        float Czu = (vz > 0.0f) ? bwdd(0, STRZ, zlo) : fwdd(0, STRZ, zhi);
        float divv = cen(1, STRX, xlo, xhi) + cen(2, STRY, ylo, yhi) + cen(3, STRZ, zlo, zhi);
        float adv = -(vx * Cxu + vy * Cyu + vz * Czu) - S[0][h] * divv;

        out[batchOff + (ix * NXX + iy) * NXX + iz] = diff + adv;
    }
}

extern "C" void kernel_launch(void* const* d_in, const int* in_sizes, int n_in,
                              void* d_out, int out_size, void* d_ws, size_t ws_size,
                              hipStream_t stream) {
    (void)in_sizes; (void)n_in; (void)out_size; (void)d_ws; (void)ws_size;
    dim3 grid(NXX / TX, NXX / TY, (NXX / TZ) * NB);   // 40 x 40 x 10
    dim3 block(NTHREADS);
    piano_energyv_kernel<<<grid, block, 0, stream>>>(
        (const float*)d_in[0], (const float*)d_in[1], (const float*)d_in[2],
        (const float*)d_in[3], (const float*)d_in[4], (const float*)d_in[5],
        (const float*)d_in[6], (const float*)d_in[7], (const float*)d_in[8],
        (const float*)d_in[9], (float*)d_out);
}